// mLSTMCell_67302137528893
// MI455X (gfx1250) — compile-verified
//
#include <hip/hip_runtime.h>
#include <cstdint>
#include <cstddef>

// Problem constants (match reference)
#define B_ 4
#define S_ 8192
#define D_ 1024
#define CHUNK 64
#define NCHUNK (S_ / CHUNK)   // 128 chunks along S
#define CTOT (B_ * D_)        // 4096 independent channels
#define TPB 256
#define TS 8                  // timesteps per async tile
#define NTILE (CHUNK / TS)    // 8 tiles per chunk

typedef __attribute__((ext_vector_type(4))) float v4f;  // native vector for NT stores

// ---------- CDNA5 async global->LDS copy (ASYNCcnt-tracked) ----------
__device__ __forceinline__ void async_copy_b128(const float4* gsrc, float4* lds_dst) {
  // Low 32 bits of the generic shared address are the wave-relative LDS byte
  // address (flat LDS aperture truncation rule, ISA 10.2).
  unsigned loff = (unsigned)(size_t)lds_dst;
  asm volatile("global_load_async_to_lds_b128 %0, %1, off"
               :: "v"(loff), "v"(gsrc)
               : "memory");
}

__device__ __forceinline__ void wait_async_le8() {
#if __has_builtin(__builtin_amdgcn_s_wait_asynccnt)
  __builtin_amdgcn_s_wait_asynccnt(8);
  asm volatile("" ::: "memory");
#else
  asm volatile("s_wait_asynccnt 0x8" ::: "memory");
#endif
}
__device__ __forceinline__ void wait_async_0() {
#if __has_builtin(__builtin_amdgcn_s_wait_asynccnt)
  __builtin_amdgcn_s_wait_asynccnt(0);
  asm volatile("" ::: "memory");
#else
  asm volatile("s_wait_asynccnt 0x0" ::: "memory");
#endif
}

// ---------- gate math: f = sigmoid(-diff), i = 1-f, v = i * g(hidden) ----------
// diff = softplus(-x*wf) - softplus(-x*wi)  =>  e^diff = (1+e^{-fg})/(1+e^{-ig})
// f = (1+Ei)/(2+Ei+Ef), i = (1+Ef)/(2+Ei+Ef); g(z) = z+0.5 (z>=0) else sigmoid(z)
__device__ __forceinline__ void gates(float x, float wf, float wi, float wh,
                                      float& f, float& v) {
  float fg = x * wf, ig = x * wi, hg = x * wh;
  float Ef = __expf(-fg);
  float Ei = __expf(-ig);
  float r  = __builtin_amdgcn_rcpf(2.0f + Ef + Ei);
  f        = (1.0f + Ei) * r;
  float i  = (1.0f + Ef) * r;
  float Eh = __expf(-hg);                       // safe: ->0 for large hg, ->inf->g=0 for very neg hg
  float gn = __builtin_amdgcn_rcpf(1.0f + Eh);  // sigmoid(hg)
  float g  = (hg >= 0.0f) ? (hg + 0.5f) : gn;
  v        = i * g;
}

// ---------- Phase 1: per-(channel, chunk) summary (F = prod f, V = scan from 0) ----------
__global__ __launch_bounds__(TPB) void summary_kernel(
    const float* __restrict__ x, const float* __restrict__ w_i,
    const float* __restrict__ w_f, const float* __restrict__ w_h,
    float* __restrict__ Fc, float* __restrict__ Vc) {
  __shared__ float4 buf[2][TS][TPB];   // 64 KB double buffer
  const int b   = blockIdx.x / NCHUNK;
  const int k   = blockIdx.x % NCHUNK;
  const int tid = threadIdx.x;
  const int d4  = tid * 4;             // 4 contiguous channels per lane (b128)
  const float4 wi4 = *(const float4*)(w_i + d4);
  const float4 wf4 = *(const float4*)(w_f + d4);
  const float4 wh4 = *(const float4*)(w_h + d4);
  const float4* gbase = (const float4*)(x + ((size_t)(b * S_ + k * CHUNK) * D_ + d4));
  const int gstep = D_ / 4;            // float4 stride between timesteps

  // prologue: prefetch tiles 0 and 1
  #pragma unroll
  for (int j = 0; j < TS; ++j) async_copy_b128(gbase + j * gstep,        &buf[0][j][tid]);
  #pragma unroll
  for (int j = 0; j < TS; ++j) async_copy_b128(gbase + (TS + j) * gstep, &buf[1][j][tid]);

  float4 h = {0.f, 0.f, 0.f, 0.f};
  float4 F = {1.f, 1.f, 1.f, 1.f};

  for (int t = 0; t < NTILE; ++t) {
    if (t + 1 < NTILE) wait_async_le8(); else wait_async_0();  // tile t landed (in-order)
    const int pb = t & 1;
    #pragma unroll
    for (int j = 0; j < TS; ++j) {
      float4 xv = buf[pb][j][tid];
      float f, v;
      gates(xv.x, wf4.x, wi4.x, wh4.x, f, v); h.x = fmaf(f, h.x, v); F.x *= f;
      gates(xv.y, wf4.y, wi4.y, wh4.y, f, v); h.y = fmaf(f, h.y, v); F.y *= f;
      gates(xv.z, wf4.z, wi4.z, wh4.z, f, v); h.z = fmaf(f, h.z, v); F.z *= f;
      gates(xv.w, wf4.w, wi4.w, wh4.w, f, v); h.w = fmaf(f, h.w, v); F.w *= f;
    }
    if (t + 2 < NTILE) {
      #pragma unroll
      for (int j = 0; j < TS; ++j)
        async_copy_b128(gbase + ((t + 2) * TS + j) * gstep, &buf[pb][j][tid]);
    }
  }
  const size_t o = (size_t)k * CTOT + (size_t)b * D_ + d4;
  *(float4*)(Fc + o) = F;
  *(float4*)(Vc + o) = h;
}

// ---------- Phase 2: scan chunk summaries along S (tiny: 4096 threads x 128) ----------
__global__ __launch_bounds__(256) void chunk_scan_kernel(
    const float* __restrict__ Fc, const float* __restrict__ Vc,
    float* __restrict__ Hi) {
  const int c = blockIdx.x * 256 + threadIdx.x;  // channel id, coalesced per k
  float carry = 0.0f;
  #pragma unroll 1
  for (int k = 0; k < NCHUNK; ++k) {
    Hi[(size_t)k * CTOT + c] = carry;
    carry = fmaf(Fc[(size_t)k * CTOT + c], carry, Vc[(size_t)k * CTOT + c]);
  }
}

// ---------- Phase 3: recompute gates, apply carry, stream h out ----------
__global__ __launch_bounds__(TPB) void apply_kernel(
    const float* __restrict__ x, const float* __restrict__ w_i,
    const float* __restrict__ w_f, const float* __restrict__ w_h,
    const float* __restrict__ Hi, float* __restrict__ out) {
  __shared__ float4 buf[2][TS][TPB];
  const int b   = blockIdx.x / NCHUNK;
  const int k   = blockIdx.x % NCHUNK;
  const int tid = threadIdx.x;
  const int d4  = tid * 4;
  const float4 wi4 = *(const float4*)(w_i + d4);
  const float4 wf4 = *(const float4*)(w_f + d4);
  const float4 wh4 = *(const float4*)(w_h + d4);
  const size_t elem0 = (size_t)(b * S_ + k * CHUNK) * D_ + d4;
  const float4* gbase = (const float4*)(x + elem0);
  v4f*          obase = (v4f*)(out + elem0);
  const int gstep = D_ / 4;

  #pragma unroll
  for (int j = 0; j < TS; ++j) async_copy_b128(gbase + j * gstep,        &buf[0][j][tid]);
  #pragma unroll
  for (int j = 0; j < TS; ++j) async_copy_b128(gbase + (TS + j) * gstep, &buf[1][j][tid]);

  float4 h = *(const float4*)(Hi + (size_t)k * CTOT + (size_t)b * D_ + d4);

  for (int t = 0; t < NTILE; ++t) {
    if (t + 1 < NTILE) wait_async_le8(); else wait_async_0();
    const int pb = t & 1;
    #pragma unroll
    for (int j = 0; j < TS; ++j) {
      float4 xv = buf[pb][j][tid];
      float f, v;
      gates(xv.x, wf4.x, wi4.x, wh4.x, f, v); h.x = fmaf(f, h.x, v);
      gates(xv.y, wf4.y, wi4.y, wh4.y, f, v); h.y = fmaf(f, h.y, v);
      gates(xv.z, wf4.z, wi4.z, wh4.z, f, v); h.z = fmaf(f, h.z, v);
      gates(xv.w, wf4.w, wi4.w, wh4.w, f, v); h.w = fmaf(f, h.w, v);
      v4f hv = {h.x, h.y, h.z, h.w};
      __builtin_nontemporal_store(hv, obase + (size_t)(t * TS + j) * gstep);
    }
    if (t + 2 < NTILE) {
      #pragma unroll
      for (int j = 0; j < TS; ++j)
        async_copy_b128(gbase + ((t + 2) * TS + j) * gstep, &buf[pb][j][tid]);
    }
  }
}

extern "C" void kernel_launch(void* const* d_in, const int* in_sizes, int n_in,
                              void* d_out, int out_size, void* d_ws, size_t ws_size,
                              hipStream_t stream) {
  // setup_inputs order: x_t, w_i, w_f, w_h (all float32)
  const float* x  = (const float*)d_in[0];
  const float* wi = (const float*)d_in[1];
  const float* wf = (const float*)d_in[2];
  const float* wh = (const float*)d_in[3];
  float* out = (float*)d_out;

  // workspace layout: F | V | Hinit, each CTOT*NCHUNK floats (6 MiB total)
  float* Fc = (float*)d_ws;
  float* Vc = Fc + (size_t)CTOT * NCHUNK;
  float* Hi = Vc + (size_t)CTOT * NCHUNK;

  summary_kernel<<<B_ * NCHUNK, TPB, 0, stream>>>(x, wi, wf, wh, Fc, Vc);
  chunk_scan_kernel<<<CTOT / 256, 256, 0, stream>>>(Fc, Vc, Hi);
  apply_kernel<<<B_ * NCHUNK, TPB, 0, stream>>>(x, wi, wf, wh, Hi, out);
}